// StrictSSM_34849364640245
// MI455X (gfx1250) — compile-verified
//
#include <hip/hip_runtime.h>

#define DIM 128
#define TILE 16

typedef float v2f __attribute__((ext_vector_type(2)));
typedef float v8f __attribute__((ext_vector_type(8)));

__device__ __forceinline__ v8f wmma_f32_k4(v2f a, v2f b, v8f c) {
  // V_WMMA_F32_16X16X4_F32: D(16x16 f32) = A(16x4 f32) * B(4x16 f32) + C
  return __builtin_amdgcn_wmma_f32_16x16x4_f32(
      /*neg_a=*/false, a, /*neg_b=*/false, b,
      /*c_mod=*/(short)0, c, /*reuse_a=*/false, /*reuse_b=*/false);
}

// Preload this lane's A-operand fragments for one 16-row stripe:
// xa[kk] = X[m0+r][4*kk + 2*half .. +1]  (ISA 32-bit 16x4 A layout)
__device__ __forceinline__ void load_xrow(v2f (&xa)[32], const float* __restrict__ xrow,
                                          int half) {
#pragma unroll
  for (int kk = 0; kk < 32; ++kk) {
    xa[kk].x = xrow[4 * kk + 2 * half];
    xa[kk].y = xrow[4 * kk + 2 * half + 1];
  }
}

// Dst = X @ Y (TRANS=false) or X @ Y^T (TRANS=true), 128x128 f32 row-major in
// LDS. X-operand fragments are pre-hoisted in registers (xa). 8 waves, each
// owns one 16-row output stripe. Call with all 256 threads active (EXEC=~0).
template <bool TRANS>
__device__ __forceinline__ void mm128(const v2f (&xa)[32], const float* __restrict__ Y,
                                      float* __restrict__ Dst, int wave, int lane) {
  const int half = lane >> 4;
  const int r    = lane & 15;
  const int m0   = wave * TILE;
  for (int tn = 0; tn < 8; ++tn) {
    const int n0 = tn * TILE;
    v8f acc = {};
#pragma unroll
    for (int kk = 0; kk < 32; ++kk) {
      const int ka = 4 * kk + 2 * half;
      v2f b;
      if (TRANS) {  // B[k][n] = Y[n][k] -> contiguous row read of Y
        b.x = Y[(n0 + r) * DIM + ka];
        b.y = Y[(n0 + r) * DIM + ka + 1];
      } else {      // column-strided read
        b.x = Y[ka * DIM + n0 + r];
        b.y = Y[(ka + 1) * DIM + n0 + r];
      }
      acc = wmma_f32_k4(xa[kk], b, acc);
    }
#pragma unroll
    for (int j = 0; j < 8; ++j)
      Dst[(m0 + j + 8 * half) * DIM + n0 + r] = acc[j];
  }
}

__global__ __launch_bounds__(256, 1)
void kalman128_kernel(const float* __restrict__ u, const float* __restrict__ yv,
                      const float* __restrict__ Ag, const float* __restrict__ Bg,
                      const float* __restrict__ Cg, const float* __restrict__ Dg,
                      const float* __restrict__ logQ, const float* __restrict__ logR,
                      float* __restrict__ out, int T) {
  __shared__ float sA[DIM * DIM];   // A (constant)
  __shared__ float sP[DIM * DIM];   // covariance (in place)
  __shared__ float sM[DIM * DIM];   // temp M = A @ P
  __shared__ float sState[DIM], sSpred[DIM];
  __shared__ float sC[DIM], sB[DIM], sQ[DIM];
  __shared__ float sV[DIM];   // P_pred @ c   (Q folded in)
  __shared__ float sW[DIM];   // c @ P_pred   (Q folded in)
  __shared__ float sScal[2];  // S, y_pred

  const int tid  = threadIdx.x;
  const int wave = tid >> 5;
  const int lane = tid & 31;
  const int half = lane >> 4;
  const int r    = lane & 15;
  const float R   = __expf(logR[0]);
  const float Dss = Dg[0];

  for (int idx = tid; idx < DIM * DIM; idx += 256) {
    sA[idx] = Ag[idx];
    sP[idx] = ((idx >> 7) == (idx & 127)) ? 1.0f : 0.0f;  // P0 = I
  }
  if (tid < DIM) {
    sState[tid] = 0.0f;
    sC[tid] = Cg[tid];
    sB[tid] = Bg[tid];
    sQ[tid] = __expf(logQ[tid]);
  }
  __syncthreads();

  // A-operand fragments of A for M = A@P: constant across all timesteps.
  v2f xaA[32];
  load_xrow(xaA, sA + (wave * TILE + r) * DIM, half);

  for (int t = 0; t < T; ++t) {
    const float u_t = u[t];
    const float y_t = yv[t];

    // state_pred = A @ state + b*u (waves 0..3, bank-skewed; overlaps WMMA)
    if (tid < DIM) {
      const int i = tid;
      const float* arow = sA + i * DIM;
      float acc = sB[i] * u_t;
      for (int jj = 0; jj < DIM; ++jj) {
        const int j = (jj + i) & (DIM - 1);  // skew: conflict-free banks
        acc += arow[j] * sState[j];
      }
      sSpred[i] = acc;
    }
    // M = A @ P
    mm128<false>(xaA, sP, sM, wave, lane);
    __syncthreads();
    // P_pred = M @ A^T (in place into sP); hoist M-row fragments first
    {
      v2f xm[32];
      load_xrow(xm, sM + (wave * TILE + r) * DIM, half);
      mm128<true>(xm, sA, sP, wave, lane);
    }
    __syncthreads();
    // v = (P_pred + Q) @ c  and  w = c @ (P_pred + Q), split across halves
    if (tid < DIM) {
      const int i = tid;
      float v = sQ[i] * sC[i];
      const float* prow = sP + i * DIM;
      for (int jj = 0; jj < DIM; ++jj) {
        const int j = (jj + i) & (DIM - 1);  // skewed row-dot
        v += prow[j] * sC[j];
      }
      sV[i] = v;
    } else {
      const int i = tid - DIM;
      float w = sQ[i] * sC[i];
      for (int j = 0; j < DIM; ++j)          // column-dot: conflict-free
        w += sC[j] * sP[j * DIM + i];
      sW[i] = w;
    }
    __syncthreads();
    // S = c.v + R ; y_pred = c.state_pred + D*u  (wave-0 shuffle reduction)
    if (wave == 0) {
      float S = 0.0f, yp = 0.0f;
      for (int j = lane; j < DIM; j += 32) {
        S  += sC[j] * sV[j];
        yp += sC[j] * sSpred[j];
      }
      for (int off = 16; off > 0; off >>= 1) {
        S  += __shfl_down(S, off, 32);
        yp += __shfl_down(yp, off, 32);
      }
      if (lane == 0) {
        S += R;
        yp += Dss * u_t;
        sScal[0] = S;
        sScal[1] = yp;
        out[t] = yp;
      }
    }
    __syncthreads();
    // Fused: K = v/S ; state = state_pred + K*(y-yp) ;
    //        P = P_wmma + diag(Q) - outer(K, w)
    {
      const float invS = 1.0f / sScal[0];
      const float yp   = sScal[1];
      if (tid < DIM)
        sState[tid] = sSpred[tid] + sV[tid] * invS * (y_t - yp);
      for (int idx = tid; idx < DIM * DIM; idx += 256) {
        const int i = idx >> 7;        // same row across a wave -> broadcast
        const int j = idx & (DIM - 1); // consecutive cols -> conflict-free
        const float k = sV[i] * invS;
        const float qadd = (i == j) ? sQ[i] : 0.0f;
        sP[idx] = sP[idx] + qadd - k * sW[j];
      }
    }
    __syncthreads();
  }
}

extern "C" void kernel_launch(void* const* d_in, const int* in_sizes, int n_in,
                              void* d_out, int out_size, void* d_ws, size_t ws_size,
                              hipStream_t stream) {
  const float* u  = (const float*)d_in[0];
  const float* y  = (const float*)d_in[1];
  const float* A  = (const float*)d_in[2];
  const float* B  = (const float*)d_in[3];
  const float* C  = (const float*)d_in[4];
  const float* D  = (const float*)d_in[5];
  const float* lq = (const float*)d_in[6];
  const float* lr = (const float*)d_in[7];
  float* out = (float*)d_out;
  const int T = in_sizes[0];
  hipLaunchKernelGGL(kalman128_kernel, dim3(1), dim3(256), 0, stream,
                     u, y, A, B, C, D, lq, lr, out, T);
}